// VQEmbedding_22780506538499
// MI455X (gfx1250) — compile-verified
//
#include <hip/hip_runtime.h>

typedef __attribute__((ext_vector_type(2))) float v2f;
typedef __attribute__((ext_vector_type(8))) float v8f;

#define EMBED_DIM       256
#define NUM_EMB         20
#define KPAD            32
#define ROWS_PER_WAVE   16
#define WAVES_PER_BLOCK 8

// ---------------------------------------------------------------------------
// Kernel 1: codebook squared norms (padded to 32) into d_ws[0..31]
// ---------------------------------------------------------------------------
__global__ void vq_prep(const float* __restrict__ emb, float* __restrict__ enorm) {
    int lane = threadIdx.x;   // 32 threads
    float s;
    if (lane < NUM_EMB) {
        const float* e = emb + lane * EMBED_DIM;
        s = 0.f;
        for (int d = 0; d < EMBED_DIM; d += 4) {
            float4 v = *(const float4*)(e + d);
            s += v.x * v.x + v.y * v.y + v.z * v.z + v.w * v.w;
        }
    } else {
        s = 3.0e38f;  // padding: never wins argmin
    }
    enorm[lane] = s;
}

// ---------------------------------------------------------------------------
// Kernel 2: main. One wave32 handles a 16-row tile of z.
//   pass 1: dot[16x32] = z_tile @ embT via v_wmma_f32_16x16x4_f32 (2 tiles)
//   argmin over 20 codes per row (shfl_xor butterfly in 16-lane halves)
//   pass 2: z_q_st = z + (e - z), accumulate (e - z)^2 deterministically
// ---------------------------------------------------------------------------
__global__ __launch_bounds__(256) void vq_main(
    const float* __restrict__ z,
    const float* __restrict__ emb,
    const float* __restrict__ enorm_g,
    float* __restrict__ zq_out,
    float* __restrict__ idx_out,
    float* __restrict__ block_partials,
    int n_tiles)
{
    __shared__ float semb[KPAD * EMBED_DIM];   // 32 KB, codes padded with zeros
    __shared__ float senorm[KPAD];
    __shared__ float swsum[WAVES_PER_BLOCK];

    const int tid = threadIdx.x;

    // cooperative load of padded codebook into LDS
    for (int e = tid; e < KPAD * EMBED_DIM; e += 256) {
        int row = e >> 8;                       // /EMBED_DIM
        semb[e] = (row < NUM_EMB) ? emb[e] : 0.f;
    }
    if (tid < KPAD) senorm[tid] = enorm_g[tid];
    __syncthreads();

    const int wave = tid >> 5;
    const int lane = tid & 31;
    const int tile = blockIdx.x * WAVES_PER_BLOCK + wave;

    float wave_sum = 0.f;

    if (tile < n_tiles) {                       // wave-uniform: EXEC all ones inside
        const int r0    = tile * ROWS_PER_WAVE;
        const int half  = lane & 15;            // M for A-frag, N for B-frag
        const int khalf = lane >> 4;
        const int kofs  = khalf * 2;            // K-pair {kofs, kofs+1}

        const float* zrow  = z + (size_t)(r0 + half) * EMBED_DIM;
        const float* brow0 = &semb[half * EMBED_DIM];          // codes 0..15
        const float* brow1 = &semb[(16 + half) * EMBED_DIM];   // codes 16..31 (pad)

        v8f acc0 = {};
        v8f acc1 = {};
        #pragma unroll 8
        for (int d = 0; d < EMBED_DIM; d += 4) {
            v2f a  = *(const v2f*)(zrow  + d + kofs);
            v2f b0 = *(const v2f*)(brow0 + d + kofs);
            v2f b1 = *(const v2f*)(brow1 + d + kofs);
            acc0 = __builtin_amdgcn_wmma_f32_16x16x4_f32(false, a, false, b0,
                                                         (short)0, acc0, false, false);
            acc1 = __builtin_amdgcn_wmma_f32_16x16x4_f32(false, a, false, b1,
                                                         (short)0, acc1, false, false);
        }

        // ---- per-row argmin of (||e||^2 - 2*dot); ||z||^2 is row-constant ----
        int idxFull[16];
        #pragma unroll
        for (int j = 0; j < 8; ++j) {
            // C/D layout: lanes 0-15 hold row M=j (N=lane), lanes 16-31 row M=j+8
            float v  = senorm[half] - 2.f * acc0[j];
            int   bi = half;
            float v1 = (half < (NUM_EMB - 16)) ? (senorm[16 + half] - 2.f * acc1[j])
                                               : 3.0e38f;
            int   i1 = 16 + half;
            if (v1 < v || (v1 == v && i1 < bi)) { v = v1; bi = i1; }
            #pragma unroll
            for (int sh = 1; sh < 16; sh <<= 1) {   // stays within 16-lane half
                float v2 = __shfl_xor(v,  sh, 32);
                int   i2 = __shfl_xor(bi, sh, 32);
                if (v2 < v || (v2 == v && i2 < bi)) { v = v2; bi = i2; }
            }
            if ((lane & 15) == 0)   // lane 0 -> row j, lane 16 -> row j+8
                idx_out[r0 + khalf * 8 + j] = (float)bi;
            idxFull[j]     = __shfl(bi, 0, 32);
            idxFull[j + 8] = __shfl(bi, 16, 32);
        }

        // ---- pass 2: straight-through output + loss partial (fixed order) ----
        float s = 0.f;
        #pragma unroll
        for (int mm = 0; mm < 16; ++mm) {
            const size_t base = (size_t)(r0 + mm) * EMBED_DIM + lane * 8;
            const float* zp = z + base;
            const float* ep = &semb[idxFull[mm] * EMBED_DIM + lane * 8];
            float*       op = zq_out + base;
            float4 z0 = *(const float4*)(zp);
            float4 z1 = *(const float4*)(zp + 4);
            float4 e0 = *(const float4*)(ep);
            float4 e1 = *(const float4*)(ep + 4);
            float4 o0, o1, d0, d1;
            d0.x = e0.x - z0.x; o0.x = z0.x + d0.x; s += d0.x * d0.x;
            d0.y = e0.y - z0.y; o0.y = z0.y + d0.y; s += d0.y * d0.y;
            d0.z = e0.z - z0.z; o0.z = z0.z + d0.z; s += d0.z * d0.z;
            d0.w = e0.w - z0.w; o0.w = z0.w + d0.w; s += d0.w * d0.w;
            d1.x = e1.x - z1.x; o1.x = z1.x + d1.x; s += d1.x * d1.x;
            d1.y = e1.y - z1.y; o1.y = z1.y + d1.y; s += d1.y * d1.y;
            d1.z = e1.z - z1.z; o1.z = z1.z + d1.z; s += d1.z * d1.z;
            d1.w = e1.w - z1.w; o1.w = z1.w + d1.w; s += d1.w * d1.w;
            *(float4*)(op)     = o0;
            *(float4*)(op + 4) = o1;
        }
        // wave reduction (fixed butterfly order -> deterministic)
        #pragma unroll
        for (int sh = 1; sh < 32; sh <<= 1) s += __shfl_xor(s, sh, 32);
        wave_sum = s;
    }

    if (lane == 0) swsum[wave] = wave_sum;
    __syncthreads();
    if (tid == 0) {
        float t = 0.f;
        #pragma unroll
        for (int w = 0; w < WAVES_PER_BLOCK; ++w) t += swsum[w];
        block_partials[blockIdx.x] = t;
    }
}

// ---------------------------------------------------------------------------
// Kernel 3: deterministic final reduction of per-block partials -> loss
// ---------------------------------------------------------------------------
__global__ void vq_finish(const float* __restrict__ partials, int nparts,
                          float* __restrict__ loss_out, float scale) {
    __shared__ float sred[256];
    int tid = threadIdx.x;
    float s = 0.f;
    for (int i = tid; i < nparts; i += 256) s += partials[i];   // fixed order
    sred[tid] = s;
    __syncthreads();
    for (int off = 128; off > 0; off >>= 1) {
        if (tid < off) sred[tid] += sred[tid + off];
        __syncthreads();
    }
    if (tid == 0) *loss_out = sred[0] * scale;
}

// ---------------------------------------------------------------------------
extern "C" void kernel_launch(void* const* d_in, const int* in_sizes, int n_in,
                              void* d_out, int out_size, void* d_ws, size_t ws_size,
                              hipStream_t stream) {
    const float* z   = (const float*)d_in[0];   // [N, 256] f32
    const float* emb = (const float*)d_in[1];   // [20, 256] f32

    const int n_rows = in_sizes[0] / EMBED_DIM;                     // 500000
    const int n_tiles = (n_rows + ROWS_PER_WAVE - 1) / ROWS_PER_WAVE;   // 31250
    const int n_blocks = (n_tiles + WAVES_PER_BLOCK - 1) / WAVES_PER_BLOCK;

    float* out   = (float*)d_out;
    float* zq    = out;                                   // [N*256]
    float* loss  = out + (size_t)n_rows * EMBED_DIM;      // [1]
    float* idxf  = loss + 1;                              // [N] indices as float

    float* enorm    = (float*)d_ws;                       // [32]
    float* partials = enorm + KPAD;                       // [n_blocks]

    vq_prep<<<1, 32, 0, stream>>>(emb, enorm);
    vq_main<<<n_blocks, 256, 0, stream>>>(z, emb, enorm, zq, idxf, partials, n_tiles);
    vq_finish<<<1, 256, 0, stream>>>(partials, n_blocks, loss,
                                     1.25f / ((float)n_rows * (float)EMBED_DIM));
}